// dotproductattention_10161892623130
// MI455X (gfx1250) — compile-verified
//
#include <hip/hip_runtime.h>
#include <hip/hip_bf16.h>

typedef float v2f __attribute__((ext_vector_type(2)));
typedef float v8f __attribute__((ext_vector_type(8)));

#define B_    2
#define L_    512
#define DIN   512
#define DOUT  256

// ---------------------------------------------------------------------------
// Fused GEMM (X[1024,512] x W[512,256]) + bias + row l2-normalize, via
// V_WMMA_F32_16X16X4_F32. One block = 16 rows x 256 cols. 8 waves/block,
// each wave owns two 16x16 N-tiles and iterates K in steps of 4.
// ---------------------------------------------------------------------------
__global__ __launch_bounds__(256) void gemm_bias_l2norm_wmma(
    const float* __restrict__ X, const float* __restrict__ W,
    const float* __restrict__ bias, float* __restrict__ Y)
{
  const int tid  = threadIdx.x;
  const int lane = tid & 31;
  const int wave = tid >> 5;
  const int hi   = lane >> 4;   // half-wave: 0 -> K=k0..k0+1, 1 -> K=k0+2..k0+3
  const int lq   = lane & 15;

  const int mtile = blockIdx.x;             // 16 rows of X per block
  const int arow  = mtile * 16 + lq;        // A-matrix row fed by this lane
  const float* Xrow = X + (size_t)arow * DIN;

  const int col0 = (wave * 2) * 16 + lq;    // two N-tiles per wave
  const int col1 = col0 + 16;

  v8f acc0 = {0.f,0.f,0.f,0.f,0.f,0.f,0.f,0.f};
  v8f acc1 = {0.f,0.f,0.f,0.f,0.f,0.f,0.f,0.f};

  for (int k0 = 0; k0 < DIN; k0 += 4) {
    const int kb = k0 + 2 * hi;
    // A 16x4 tile: VGPR0 = X[row, kb], VGPR1 = X[row, kb+1]  (b64 load)
    v2f a = *reinterpret_cast<const v2f*>(Xrow + kb);
    // B 4x16 tiles: VGPR0 = W[kb, col], VGPR1 = W[kb+1, col]
    v2f b0, b1;
    b0.x = W[(size_t)kb * DOUT + col0];
    b0.y = W[(size_t)(kb + 1) * DOUT + col0];
    b1.x = W[(size_t)kb * DOUT + col1];
    b1.y = W[(size_t)(kb + 1) * DOUT + col1];
    acc0 = __builtin_amdgcn_wmma_f32_16x16x4_f32(false, a, false, b0,
                                                 (short)0, acc0, false, false);
    acc1 = __builtin_amdgcn_wmma_f32_16x16x4_f32(false, a, false, b1,
                                                 (short)0, acc1, false, false);
  }

  const float bz0 = bias[col0];
  const float bz1 = bias[col1];

  __shared__ float s_sum[16];
  if (tid < 16) s_sum[tid] = 0.f;
  __syncthreads();

  // C/D layout: VGPR r, lanes 0-15 -> row r; lanes 16-31 -> row r+8
#pragma unroll
  for (int r = 0; r < 8; ++r) {
    acc0[r] += bz0;
    acc1[r] += bz1;
    float p = acc0[r] * acc0[r] + acc1[r] * acc1[r];
    // reduce across the 16 lanes of this half-wave (xor 1,2,4,8 stay in-half)
    p += __shfl_xor(p, 1, 32);
    p += __shfl_xor(p, 2, 32);
    p += __shfl_xor(p, 4, 32);
    p += __shfl_xor(p, 8, 32);
    if (lq == 0) atomicAdd(&s_sum[r + 8 * hi], p);
  }
  __syncthreads();

#pragma unroll
  for (int r = 0; r < 8; ++r) {
    const int rr = r + 8 * hi;
    const float sc = rsqrtf(fmaxf(s_sum[rr], 1e-12f));
    const size_t gr = (size_t)(mtile * 16 + rr) * DOUT;
    Y[gr + col0] = acc0[r] * sc;
    Y[gr + col1] = acc1[r] * sc;
  }
}

// ---------------------------------------------------------------------------
// scores[b,i,j] = dot(q[b,j], normalize(q[b,i] + rel[b,i,j]))
// One wave handles (b,i) and 8 consecutive j. rel is streamed once (512 MiB,
// HBM-bound); q stays hot in L2. b128 loads + prefetch, wave32 shfl reduce.
// ---------------------------------------------------------------------------
__global__ __launch_bounds__(256) void scores_fused(
    const float* __restrict__ q, const float* __restrict__ rel,
    float* __restrict__ scores)
{
  const int lane = threadIdx.x & 31;
  const int gw   = blockIdx.x * 8 + (threadIdx.x >> 5);
  const int pair0 = gw * 8;                // 8 consecutive j share (b,i)
  const int b  = pair0 / (L_ * L_);
  const int rm = pair0 % (L_ * L_);
  const int i  = rm / L_;
  const int j0 = rm % L_;

  const int e = lane * 4;                  // two float4 per lane: e, e+128
  const float* qi = q + (size_t)(b * L_ + i) * DOUT;
  const float4 qa = *reinterpret_cast<const float4*>(qi + e);
  const float4 qb = *reinterpret_cast<const float4*>(qi + 128 + e);

  const float* rp = rel + ((size_t)(b * L_ + i) * L_ + j0) * DOUT;
  float* sp = scores + (size_t)(b * L_ + i) * L_ + j0;

  for (int jj = 0; jj < 8; ++jj) {
    if (jj < 7) {
      __builtin_prefetch(rp + DOUT + e, 0, 0);
      __builtin_prefetch(rp + DOUT + 128 + e, 0, 0);
    }
    const float4 ra = *reinterpret_cast<const float4*>(rp + e);
    const float4 rb = *reinterpret_cast<const float4*>(rp + 128 + e);
    const float* qj = q + (size_t)(b * L_ + j0 + jj) * DOUT;
    const float4 ja = *reinterpret_cast<const float4*>(qj + e);
    const float4 jb = *reinterpret_cast<const float4*>(qj + 128 + e);

    const float4 ta = {qa.x + ra.x, qa.y + ra.y, qa.z + ra.z, qa.w + ra.w};
    const float4 tb = {qb.x + rb.x, qb.y + rb.y, qb.z + rb.z, qb.w + rb.w};

    float ss = ta.x * ta.x + ta.y * ta.y + ta.z * ta.z + ta.w * ta.w
             + tb.x * tb.x + tb.y * tb.y + tb.z * tb.z + tb.w * tb.w;
    float dd = ta.x * ja.x + ta.y * ja.y + ta.z * ja.z + ta.w * ja.w
             + tb.x * jb.x + tb.y * jb.y + tb.z * jb.z + tb.w * jb.w;

    ss += __shfl_xor(ss, 1, 32);  dd += __shfl_xor(dd, 1, 32);
    ss += __shfl_xor(ss, 2, 32);  dd += __shfl_xor(dd, 2, 32);
    ss += __shfl_xor(ss, 4, 32);  dd += __shfl_xor(dd, 4, 32);
    ss += __shfl_xor(ss, 8, 32);  dd += __shfl_xor(dd, 8, 32);
    ss += __shfl_xor(ss, 16, 32); dd += __shfl_xor(dd, 16, 32);

    if (lane == 0) sp[jj] = dd * rsqrtf(fmaxf(ss, 1e-12f));
    rp += DOUT;
  }
}

extern "C" void kernel_launch(void* const* d_in, const int* in_sizes, int n_in,
                              void* d_out, int out_size, void* d_ws, size_t ws_size,
                              hipStream_t stream) {
  (void)in_sizes; (void)n_in; (void)out_size; (void)d_ws; (void)ws_size;
  const float* queries = (const float*)d_in[0];
  // d_in[1] = keys  : unused (k is deleted in the reference)
  const float* values  = (const float*)d_in[2];
  const float* rel     = (const float*)d_in[3];
  const float* Wq      = (const float*)d_in[4];
  const float* bq      = (const float*)d_in[5];
  // d_in[6], d_in[7] = Wk, bk : unused
  const float* Wv      = (const float*)d_in[8];
  const float* bv      = (const float*)d_in[9];

  float* out    = (float*)d_out;
  float* scores = out;                         // B*L*L     = 524288
  float* v_out  = out + (size_t)B_ * L_ * L_;  // B*L*DOUT  = 262144
  float* q_out  = v_out + (size_t)B_ * L_ * DOUT;

  const int mtiles = (B_ * L_) / 16;           // 64 blocks
  gemm_bias_l2norm_wmma<<<mtiles, 256, 0, stream>>>(queries, Wq, bq, q_out);
  gemm_bias_l2norm_wmma<<<mtiles, 256, 0, stream>>>(values,  Wv, bv, v_out);

  const int sblocks = (B_ * L_ * L_) / 64;     // 8 waves * 8 j per block
  scores_fused<<<sblocks, 256, 0, stream>>>(q_out, rel, scores);
}